// GATv2_87256555585738
// MI455X (gfx1250) — compile-verified
//
#include <hip/hip_runtime.h>
#include <math.h>

// ---------------------------------------------------------------------------
// GATv2 (2-layer) forward for MI455X / gfx1250.
//   out  = mean-head( layer2( relu( layer1(x) ) ) )      [N, 64]
//   a1   = layer-1 edge attention coefficients            [E+N, 2]
//   a2   = layer-2 edge attention coefficients            [E+N, 2]
// GEMMs use V_WMMA_F32_16X16X4_F32 (exact fp32 matrix pipe); edge phase is
// the bandwidth bottleneck and uses float4 gathers + f32 atomics.
// ---------------------------------------------------------------------------

typedef __attribute__((ext_vector_type(2))) float v2f;
typedef __attribute__((ext_vector_type(8))) float v8f;

#define NEG_SLOPE 0.2f

// Order-preserving float <-> uint mapping so atomicMax(u32) implements
// a float segment-max (init value 0u is below every encoded float).
__device__ __forceinline__ unsigned encOrd(float f) {
  unsigned u = __float_as_uint(f);
  return (u & 0x80000000u) ? ~u : (u | 0x80000000u);
}
__device__ __forceinline__ float decOrd(unsigned u) {
  return (u & 0x80000000u) ? __uint_as_float(u ^ 0x80000000u)
                           : __uint_as_float(~u);
}

__device__ __forceinline__ float lrelu(float v) {
  return v >= 0.0f ? v : v * NEG_SLOPE;
}

// ---------------------------------------------------------------------------
// D[M, Out] = A[M, K] @ W[Out, K]^T  via V_WMMA_F32_16X16X4_F32.
// One wave per 16x16 output tile; 8 waves per 256-thread block.
//
// f32 A-matrix 16x4 layout (ISA 7.12.2): lanes 0-15 / 16-31 both hold M=lane&15;
// VGPR0 = {K=0 | K=2}, VGPR1 = {K=1 | K=3}  -> per-lane contiguous float2 at
// column (k + 2*(lane>>4)). B (= W^T, 4x16) mirrors this with N=lane&15.
// C/D: VGPR r -> row r (lanes 0-15) / r+8 (lanes 16-31), col = lane&15.
// ---------------------------------------------------------------------------
__global__ __launch_bounds__(256) void k_gemm_f32_wmma(
    const float* __restrict__ A,   // [M, K] row-major
    const float* __restrict__ W,   // [Out, K] row-major
    float* __restrict__ D,         // [M, Out] row-major
    int M, int K, int Out)
{
  const int lane = threadIdx.x & 31;
  const int wid  = threadIdx.x >> 5;
  const int NT   = Out >> 4;
  const int MT   = (M + 15) >> 4;
  const int tile = blockIdx.x * 8 + wid;
  if (tile >= MT * NT) return;                    // wave-uniform: EXEC all-1s
  const int mtile = tile / NT;
  const int ntile = tile - mtile * NT;

  const int lid  = lane & 15;
  const int half = lane >> 4;
  const int koff = half * 2;

  int m_row = mtile * 16 + lid;
  if (m_row >= M) m_row = M - 1;                  // clamp (M=50000 is exact)
  const int n_col = ntile * 16 + lid;

  const float* aRow = A + (size_t)m_row * K + koff;
  const float* bRow = W + (size_t)n_col * K + koff;

  v8f c = {};
  for (int k = 0; k < K; k += 4) {
    v2f a = *(const v2f*)(aRow + k);
    v2f b = *(const v2f*)(bRow + k);
    c = __builtin_amdgcn_wmma_f32_16x16x4_f32(
        /*neg_a=*/false, a, /*neg_b=*/false, b,
        /*c_mod=*/(short)0, c, /*reuse_a=*/false, /*reuse_b=*/false);
  }

#pragma unroll
  for (int r = 0; r < 8; ++r) {
    const int m_out = mtile * 16 + r + 8 * half;
    if (m_out < M)
      D[(size_t)m_out * Out + n_col] = c[r];
  }
}

// ---------------------------------------------------------------------------
// Pass A: per-edge logits   logit[e,h] = att[h] . lrelu(h[src]+h[dst])
// One wave per edge. Lane covers 4 channels (float4); lanes 0-15 = head 0,
// lanes 16-31 = head 1. 16-lane xor-reduce per head. Writes logit into the
// output alpha slot (reused as scratch) and atomicMax into per-(dst,head) max.
// ---------------------------------------------------------------------------
__global__ __launch_bounds__(256) void k_edge_logits(
    const float* __restrict__ h,      // [N, 128]
    const int*  __restrict__ srcA,
    const int*  __restrict__ dstA,
    int E, int Etot,
    const float* __restrict__ att,    // [2, 64] flat = 128 floats
    float* __restrict__ logitOut,     // [Etot, 2]
    unsigned* __restrict__ mEnc)      // [N, 2] encoded max
{
  const int e    = (int)((blockIdx.x * blockDim.x + threadIdx.x) >> 5);
  const int lane = threadIdx.x & 31;
  if (e >= Etot) return;

  int s, d;
  if (e < E) { s = srcA[e]; d = dstA[e]; } else { s = e - E; d = s; }

  const float4 hs = ((const float4*)h)[(size_t)s * 32 + lane];
  const float4 hd = ((const float4*)h)[(size_t)d * 32 + lane];
  const float4 at = ((const float4*)att)[lane];

  float sum = at.x * lrelu(hs.x + hd.x)
            + at.y * lrelu(hs.y + hd.y)
            + at.z * lrelu(hs.z + hd.z)
            + at.w * lrelu(hs.w + hd.w);

  sum += __shfl_xor(sum, 8);
  sum += __shfl_xor(sum, 4);
  sum += __shfl_xor(sum, 2);
  sum += __shfl_xor(sum, 1);

  if ((lane & 15) == 0) {
    const int head = lane >> 4;
    logitOut[(size_t)e * 2 + head] = sum;
    atomicMax(&mEnc[(size_t)d * 2 + head], encOrd(sum));
  }
}

// ---------------------------------------------------------------------------
// Pass B: p = exp(logit - max[dst]); denom[dst] += p. One thread per (e,head).
// ---------------------------------------------------------------------------
__global__ __launch_bounds__(256) void k_exp_denom(
    const int* __restrict__ dstA, int E, int Etot,
    float* __restrict__ pbuf,            // in: logit, out: p
    const unsigned* __restrict__ mEnc,
    float* __restrict__ denom)           // [N, 2]
{
  const int idx = blockIdx.x * blockDim.x + threadIdx.x;
  if (idx >= 2 * Etot) return;
  const int e  = idx >> 1;
  const int hh = idx & 1;
  const int d  = (e < E) ? dstA[e] : (e - E);
  const float mv = decOrd(mEnc[(size_t)d * 2 + hh]);
  const float p  = expf(pbuf[idx] - mv);
  pbuf[idx] = p;
  atomicAdd(&denom[(size_t)d * 2 + hh], p);
}

// ---------------------------------------------------------------------------
// Pass C: alpha = p / denom[dst]; agg[dst] += alpha * h[src].
// One wave per edge; lane handles 4 channels of its head. Lanes 0/16 write
// the final alpha back to the output buffer.
// ---------------------------------------------------------------------------
__global__ __launch_bounds__(256) void k_aggregate(
    const float* __restrict__ h,
    const int* __restrict__ srcA,
    const int* __restrict__ dstA,
    int E, int Etot,
    float* __restrict__ pbuf,            // in: p, out: alpha
    const float* __restrict__ denom,
    float* __restrict__ agg)             // [N, 128] zero-initialized
{
  const int e    = (int)((blockIdx.x * blockDim.x + threadIdx.x) >> 5);
  const int lane = threadIdx.x & 31;
  if (e >= Etot) return;

  int s, d;
  if (e < E) { s = srcA[e]; d = dstA[e]; } else { s = e - E; d = s; }

  const int head = lane >> 4;
  const float p    = pbuf[(size_t)e * 2 + head];
  const float den  = denom[(size_t)d * 2 + head];
  const float alpha = p / den;
  if ((lane & 15) == 0)
    pbuf[(size_t)e * 2 + head] = alpha;

  const float4 hs = ((const float4*)h)[(size_t)s * 32 + lane];
  float* ag = agg + (size_t)d * 128 + lane * 4;
  atomicAdd(ag + 0, alpha * hs.x);
  atomicAdd(ag + 1, alpha * hs.y);
  atomicAdd(ag + 2, alpha * hs.z);
  atomicAdd(ag + 3, alpha * hs.w);
}

__global__ __launch_bounds__(256) void k_relu(float* __restrict__ v, int n) {
  const int i = blockIdx.x * blockDim.x + threadIdx.x;
  if (i < n) v[i] = fmaxf(v[i], 0.0f);
}

// out[n,c] = 0.5 * (agg[n,c] + agg[n,64+c])   (mean over 2 heads)
__global__ __launch_bounds__(256) void k_mean_heads(
    const float* __restrict__ agg, float* __restrict__ out, int N)
{
  const int i = blockIdx.x * blockDim.x + threadIdx.x;
  if (i >= N * 64) return;
  const int n = i >> 6, cc = i & 63;
  out[i] = 0.5f * (agg[(size_t)n * 128 + cc] + agg[(size_t)n * 128 + 64 + cc]);
}

// ---------------------------------------------------------------------------
extern "C" void kernel_launch(void* const* d_in, const int* in_sizes, int n_in,
                              void* d_out, int out_size, void* d_ws, size_t ws_size,
                              hipStream_t stream) {
  const float* x    = (const float*)d_in[0];
  const int*   ei   = (const int*)  d_in[1];
  // d_in[2] = getAttnCoef (always emit attention coefficients)
  const float* W1   = (const float*)d_in[3];
  const float* att1 = (const float*)d_in[4];
  const float* W2   = (const float*)d_in[5];
  const float* att2 = (const float*)d_in[6];

  const int DIN = 64, HC = 128, H = 2, C = 64;
  const int N    = in_sizes[0] / DIN;     // 50000
  const int E    = in_sizes[1] / 2;       // 1,600,000
  const int Etot = E + N;                 // + self loops
  const int* srcA = ei;
  const int* dstA = ei + E;

  float* out = (float*)d_out;                    // [N, 64]
  float* a1  = out + (size_t)N * C;              // [Etot, 2]
  float* a2  = a1 + (size_t)Etot * H;            // [Etot, 2]

  // workspace layout: bufA | bufB | mEnc | denom  (bufB..denom contiguous)
  const size_t S = (size_t)N * HC * sizeof(float);
  char* ws = (char*)d_ws;
  float*    bufA  = (float*)ws;                          // h (projected)
  float*    bufB  = (float*)(ws + S);                    // agg / relu(h1)
  unsigned* mEnc  = (unsigned*)(ws + 2 * S);             // [N,2]
  float*    denom = (float*)(ws + 2 * S + (size_t)N * H * sizeof(unsigned));
  const size_t zbytes = S + (size_t)N * H * (sizeof(unsigned) + sizeof(float));

  const dim3 blk(256);
  const int gemmBlocks = (((N + 15) / 16) * (HC / 16) + 7) / 8;  // 8 waves/blk
  const int edgeBlocks = (Etot + 7) / 8;                         // 1 wave/edge
  const int ehBlocks   = (2 * Etot + 255) / 256;
  const int reluBlocks = (N * HC + 255) / 256;
  const int meanBlocks = (N * C + 255) / 256;

  // ---------------- layer 1 ----------------
  hipMemsetAsync(bufB, 0, zbytes, stream);                       // agg1, m, denom
  k_gemm_f32_wmma<<<gemmBlocks, blk, 0, stream>>>(x, W1, bufA, N, DIN, HC);
  k_edge_logits  <<<edgeBlocks, blk, 0, stream>>>(bufA, srcA, dstA, E, Etot, att1, a1, mEnc);
  k_exp_denom    <<<ehBlocks,   blk, 0, stream>>>(dstA, E, Etot, a1, mEnc, denom);
  k_aggregate    <<<edgeBlocks, blk, 0, stream>>>(bufA, srcA, dstA, E, Etot, a1, denom, bufB);
  k_relu         <<<reluBlocks, blk, 0, stream>>>(bufB, N * HC);

  // ---------------- layer 2 ----------------
  k_gemm_f32_wmma<<<gemmBlocks, blk, 0, stream>>>(bufB, W2, bufA, N, HC, HC);
  hipMemsetAsync(bufB, 0, zbytes, stream);                       // agg2, m, denom
  k_edge_logits  <<<edgeBlocks, blk, 0, stream>>>(bufA, srcA, dstA, E, Etot, att2, a2, mEnc);
  k_exp_denom    <<<ehBlocks,   blk, 0, stream>>>(dstA, E, Etot, a2, mEnc, denom);
  k_aggregate    <<<edgeBlocks, blk, 0, stream>>>(bufA, srcA, dstA, E, Etot, a2, denom, bufB);
  k_mean_heads   <<<meanBlocks, blk, 0, stream>>>(bufB, out, N);
}